// _DifferentiableOrthogonalMatchingPursuit_40432822125178
// MI455X (gfx1250) — compile-verified
//
#include <hip/hip_runtime.h>
#include <math.h>

typedef float v2f __attribute__((ext_vector_type(2)));
typedef float v8f __attribute__((ext_vector_type(8)));

#define Bsz      32
#define Lsz      512
#define Asz      512
#define NITER    8
#define NTHREADS 256
#define NWAVES   8   // 256 threads / wave32

__global__ __launch_bounds__(NTHREADS)
void omp_hard1_kernel(const float* __restrict__ X,
                      const float* __restrict__ y,
                      float* __restrict__ out)
{
    const int b    = blockIdx.x;
    const int tid  = threadIdx.x;
    const int lane = tid & 31;
    const int wave = tid >> 5;

    __shared__ float res[Lsz];                 // residuals
    __shared__ float yb[Lsz];                  // y[b]
    __shared__ float proj[Asz];                // projections
    __shared__ float colCache[NITER][Lsz];     // selected dictionary columns of X[b]
    __shared__ float gvec[NITER];              // col_j . y
    __shared__ float Gmat[NITER][NITER];       // col_j . col_m
    __shared__ int   selIdx[NITER];
    __shared__ float selS[NITER];              // multiplicities (sum_collector values)
    __shared__ float coef[NITER];              // s_j * w_j for residual update
    __shared__ float wsol[NITER];              // solved weights
    __shared__ int   kUnique;
    __shared__ int   chosenIdx;
    __shared__ int   newAtom;
    __shared__ float redVal[NTHREADS];
    __shared__ int   redIdx[NTHREADS];

    const float* Xb   = X + (size_t)b * Lsz * Asz;
    const float* dict = X;                     // dict_ = X[0]
    const float* yrow = y + (size_t)b * Lsz;

    for (int l = tid; l < Lsz; l += NTHREADS) {
        float v = yrow[l];
        yb[l]  = v;
        res[l] = v;                            // initial residuals = y
    }
    if (tid == 0) kUnique = 0;
    __syncthreads();

    const float reg = log1pf(expf(-5.0f));     // softplus(LAMBDA_INIT)

    for (int it = 0; it < NITER; ++it) {
        // ---------- Phase 1: proj[a] = sum_l res[l] * dict[l][a]   (fp32 WMMA 16x16x4)
        {
            const int half = lane >> 4;        // K half: 0 -> K{0,1}, 1 -> K{2,3}
            const int lm   = lane & 15;
            for (int tile = wave; tile < Asz / 16; tile += NWAVES) {
                const int n0 = tile * 16;
                v8f c = {};
                for (int k0 = 0; k0 < Lsz; k0 += 4) {
                    const int kA = k0 + 2 * half;
                    float r0 = res[kA], r1 = res[kA + 1];
                    v2f a, bop;
                    a.x = (lm == 0) ? r0 : 0.0f;   // residual in row M=0 only
                    a.y = (lm == 0) ? r1 : 0.0f;
                    bop.x = dict[(size_t)kA       * Asz + n0 + lm];
                    bop.y = dict[(size_t)(kA + 1) * Asz + n0 + lm];
                    c = __builtin_amdgcn_wmma_f32_16x16x4_f32(
                            false, a, false, bop, (short)0, c, false, false);
                }
                if (lane < 16) proj[n0 + lane] = c[0];   // D row M=0: VGPR0, lanes 0..15
            }
        }
        __syncthreads();

        // ---------- Phase 2: argmax |proj| (lowest index wins ties, like jnp.argmax)
        {
            float bestV = -1.0f; int bestI = 0;
            for (int a = tid; a < Asz; a += NTHREADS) {
                float v = fabsf(proj[a]);
                if (v > bestV) { bestV = v; bestI = a; }
            }
            redVal[tid] = bestV; redIdx[tid] = bestI;
            __syncthreads();
            for (int s = NTHREADS / 2; s > 0; s >>= 1) {
                if (tid < s) {
                    float v = redVal[tid + s]; int i = redIdx[tid + s];
                    if (v > redVal[tid] || (v == redVal[tid] && i < redIdx[tid])) {
                        redVal[tid] = v; redIdx[tid] = i;
                    }
                }
                __syncthreads();
            }
            if (tid == 0) chosenIdx = redIdx[0];
        }
        __syncthreads();

        // ---------- Phase 3: selection bookkeeping (sum_collector multiplicities)
        if (tid == 0) {
            int idx = chosenIdx;
            int k = kUnique, found = -1;
            for (int j = 0; j < k; ++j) if (selIdx[j] == idx) { found = j; break; }
            if (found >= 0) { selS[found] += 1.0f; newAtom = 0; }
            else { selIdx[k] = idx; selS[k] = 1.0f; kUnique = k + 1; newAtom = 1; }
        }
        __syncthreads();

        const int k = kUnique;

        // ---------- Phase 4: new atom -> cache column, extend Gram / gvec
        if (newAtom) {
            const int j = k - 1;
            const int idx = selIdx[j];
            for (int l = tid; l < Lsz; l += NTHREADS)
                colCache[j][l] = Xb[(size_t)l * Asz + idx];
            __syncthreads();
            // tasks 0..j : G[j][m];  task k : gvec[j]   (one wave per dot product)
            for (int task = wave; task <= k; task += NWAVES) {
                float s = 0.0f;
                const float* cj = colCache[j];
                if (task < k) {
                    const float* cm = colCache[task];
                    for (int l = lane; l < Lsz; l += 32) s += cj[l] * cm[l];
                } else {
                    for (int l = lane; l < Lsz; l += 32) s += cj[l] * yb[l];
                }
                for (int m = 16; m >= 1; m >>= 1) s += __shfl_xor(s, m, 32);
                if (lane == 0) {
                    if (task < k) { Gmat[j][task] = s; Gmat[task][j] = s; }
                    else gvec[j] = s;
                }
            }
        }
        __syncthreads();

        // ---------- Phase 5: build + solve k x k ridge system (thread 0; k <= 8)
        if (tid == 0) {
            float M[NITER][NITER + 1];
            for (int r = 0; r < k; ++r) {
                for (int c2 = 0; c2 < k; ++c2)
                    M[r][c2] = selS[r] * selS[c2] * Gmat[r][c2] + (r == c2 ? reg : 0.0f);
                M[r][k] = selS[r] * gvec[r];
            }
            for (int p = 0; p < k; ++p) {                 // partial-pivot LU
                int pr = p; float pv = fabsf(M[p][p]);
                for (int r = p + 1; r < k; ++r) {
                    float v = fabsf(M[r][p]);
                    if (v > pv) { pv = v; pr = r; }
                }
                if (pr != p)
                    for (int c2 = p; c2 <= k; ++c2) {
                        float t = M[p][c2]; M[p][c2] = M[pr][c2]; M[pr][c2] = t;
                    }
                float inv = 1.0f / M[p][p];
                for (int r = p + 1; r < k; ++r) {
                    float f = M[r][p] * inv;
                    for (int c2 = p; c2 <= k; ++c2) M[r][c2] -= f * M[p][c2];
                }
            }
            for (int r = k - 1; r >= 0; --r) {
                float s = M[r][k];
                for (int c2 = r + 1; c2 < k; ++c2) s -= M[r][c2] * wsol[c2];
                wsol[r] = s / M[r][r];
            }
            for (int j = 0; j < k; ++j) coef[j] = selS[j] * wsol[j];
        }
        __syncthreads();

        // ---------- Phase 6: residuals = y - selected_D @ w  (skip final, unused)
        if (it < NITER - 1) {
            for (int l = tid; l < Lsz; l += NTHREADS) {
                float s = yb[l];
                for (int j = 0; j < k; ++j) s -= coef[j] * colCache[j][l];
                res[l] = s;
            }
            __syncthreads();
        }
    }

    // ---------- Output: W[b, a] = w_j if a selected else 0
    float* outb = out + (size_t)b * Asz;
    const int kf = kUnique;
    for (int a = tid; a < Asz; a += NTHREADS) {
        float v = 0.0f;
        for (int j = 0; j < kf; ++j) if (selIdx[j] == a) v = wsol[j];
        outb[a] = v;
    }
}

extern "C" void kernel_launch(void* const* d_in, const int* in_sizes, int n_in,
                              void* d_out, int out_size, void* d_ws, size_t ws_size,
                              hipStream_t stream) {
    (void)in_sizes; (void)n_in; (void)d_ws; (void)ws_size; (void)out_size;
    const float* X = (const float*)d_in[0];
    const float* y = (const float*)d_in[1];
    float* out = (float*)d_out;
    hipLaunchKernelGGL(omp_hard1_kernel, dim3(Bsz), dim3(NTHREADS), 0, stream, X, y, out);
}